// GCNNet2_7129645711575
// MI455X (gfx1250) — compile-verified
//
#include <hip/hip_runtime.h>
#include <hip/hip_bf16.h>

// ---------------------------------------------------------------------------
// GCNNet2 forward for MI455X (gfx1250, wave32).
// Memory/atomic-bound workload (~8-9 GB traffic vs 16.4 GFLOP); GEMMs run on
// V_WMMA_F32_16X16X4_F32 (f32-exact), A tiles staged via the Tensor Data
// Mover (tensor_load_to_lds) with double-buffered LDS and one-tile-ahead
// prefetch. TDM pad feature (interval=128 DW, amount=4 DW) reproduces the
// 132-float padded row stride used for conflict-free LDS reads.
// ---------------------------------------------------------------------------

typedef __attribute__((ext_vector_type(2))) float v2f;
typedef __attribute__((ext_vector_type(8))) float v8f;
typedef __attribute__((ext_vector_type(4))) unsigned int v4u_t;
typedef __attribute__((ext_vector_type(8))) int v8i_t;
typedef __attribute__((ext_vector_type(4))) int v4i_t;

#define D_DIM 128
#define N_CLS 10
#define BN_EPS 1e-5f

#if defined(__HIP_DEVICE_COMPILE__) && \
    __has_builtin(__builtin_amdgcn_tensor_load_to_lds) && \
    __has_builtin(__builtin_amdgcn_s_wait_tensorcnt)
#define USE_TDM 1
#else
#define USE_TDM 0
#endif

// ---------------------------------------------------------------------------
// utility kernels
// ---------------------------------------------------------------------------
__global__ void fill_zero_f32(float* __restrict__ p, long long n) {
    long long i = (long long)blockIdx.x * blockDim.x + threadIdx.x;
    if (i < n) p[i] = 0.0f;
}

__global__ void count_deg(const int* __restrict__ src, const int* __restrict__ dst,
                          float* __restrict__ dout, float* __restrict__ din, int nEdges) {
    int e = blockIdx.x * blockDim.x + threadIdx.x;
    if (e >= nEdges) return;
    unsafeAtomicAdd(&dout[src[e]], 1.0f);
    unsafeAtomicAdd(&din[dst[e]], 1.0f);
}

__global__ void inv_sqrt_deg(float* __restrict__ dout, float* __restrict__ din, int n) {
    int i = blockIdx.x * blockDim.x + threadIdx.x;
    if (i >= n) return;
    dout[i] = rsqrtf(fmaxf(dout[i], 1.0f));
    din[i]  = rsqrtf(fmaxf(din[i], 1.0f));
}

// ---------------------------------------------------------------------------
// msg = (h * inv_sqrt_out)[src];  agg[dst] += msg      (32 lanes per edge)
// ---------------------------------------------------------------------------
__global__ void gather_scatter(const float* __restrict__ h,
                               const float* __restrict__ inv_out,
                               const int* __restrict__ src,
                               const int* __restrict__ dst,
                               float* __restrict__ agg, int nEdges) {
    long long tid = (long long)blockIdx.x * blockDim.x + threadIdx.x;
    int e = (int)(tid >> 5);
    if (e >= nEdges) return;
    int lane = (int)(tid & 31);
    int s = src[e], d = dst[e];
    float sc = inv_out[s];
    const float4 v = *reinterpret_cast<const float4*>(h + (size_t)s * D_DIM + lane * 4);
    float* o = agg + (size_t)d * D_DIM + lane * 4;
    unsafeAtomicAdd(o + 0, v.x * sc);
    unsafeAtomicAdd(o + 1, v.y * sc);
    unsafeAtomicAdd(o + 2, v.z * sc);
    unsafeAtomicAdd(o + 3, v.w * sc);
}

// ---------------------------------------------------------------------------
// TDM: DMA one 16x128 f32 tile (row-major, contiguous) into LDS at ldsOff,
// inserting 4 DWORDs of padding after every 128 DWORDs -> row stride 132 f32.
// D# group0/group1 bitfields per CDNA5 ISA §8.3/8.4; groups 2/3 describe a
// degenerate (size-1) higher-dim tensor.
// ---------------------------------------------------------------------------
#if USE_TDM
__device__ __forceinline__ void tdm_load_tile(unsigned ldsOff, unsigned long long ga) {
    v4u_t g0;
    g0[0] = 1u;                                      // count=1, no gather, user desc
    g0[1] = ldsOff;                                  // lds_addr [63:32]
    g0[2] = (unsigned)(ga & 0xffffffffu);            // global_addr [95:64]
    g0[3] = (unsigned)((ga >> 32) & 0x01ffffffu)     // global_addr [120:96]
            | (2u << 30);                            // type=2 ("image") [127:126]
    v8i_t g1;
    g1[0] = (int)((2u << 16)                         // data_size = 4B
                  | (1u << 20)                       // pad_enable
                  | (6u << 22)                       // pad_interval: 2^(6+1)=128 DW
                  | (3u << 25));                     // pad_amount: 3+1 = 4 DW
    g1[1] = (int)(128u << 16);                       // tensor_dim0 = 128   [79:48]
    g1[2] = (int)(16u << 16);                        // tensor_dim1 = 16    [111:80]
    g1[3] = (int)(128u << 16);                       // tile_dim0 = 128     [127:112]
    g1[4] = 16;                                      // tile_dim1 = 16      [143:128]
    g1[5] = 128;                                     // tensor_dim0_stride  [207:160]
    g1[6] = 0;
    g1[7] = 0;
    v4i_t g2; g2[0] = 1; g2[1] = 1; g2[2] = 0; g2[3] = 0;   // tensor_dim2/3 = 1
    v4i_t g3; g3[0] = 0; g3[1] = (int)(1u << 16); g3[2] = 0; g3[3] = 0; // tensor_dim4 = 1
#if __clang_major__ >= 23
    v8i_t z8 = {0, 0, 0, 0, 0, 0, 0, 0};
    __builtin_amdgcn_tensor_load_to_lds(g0, g1, g2, g3, z8, 0);
#else
    __builtin_amdgcn_tensor_load_to_lds(g0, g1, g2, g3, 0);
#endif
}
#endif

// ---------------------------------------------------------------------------
// out[m][n] = (sum_k A[m][k]*W[k][n]) * rs[m] + bias[n], then * ps[m]
// (row scale commutes through the GEMM: diag(rs)*A*W = diag(rs)*(A*W), so it
// is applied in the epilogue and A staging is a pure copy -> TDM-friendly.)
// Persistent blocks of 8 waves; each wave owns a 16-col B slab in 64 VGPRs,
// reused across all M tiles. 32 x v_wmma_f32_16x16x4_f32 per 16x16 tile.
// nRows % 16 == 0.
// ---------------------------------------------------------------------------
__global__ void gemm128_wmma(const float* __restrict__ A,
                             const float* __restrict__ W,
                             const float* __restrict__ bias,
                             const float* __restrict__ rowscale,   // may be null
                             const float* __restrict__ postscale,  // may be null
                             float* __restrict__ out, int nRows) {
    __shared__ float Atile[2][16][132];              // pad stride 132 -> no bank conflicts
    const int lane    = threadIdx.x & 31;
    const int wave    = threadIdx.x >> 5;            // 0..7 -> N tile
    const int n0      = wave << 4;
    const int halfSel = lane >> 4;                   // 0: lanes 0-15, 1: lanes 16-31
    const int lane16  = lane & 15;
    const int col     = n0 + lane16;

    // B slab: 4x16 chunk kk -> VGPR pair. ISA layout: vgpr0 holds K=k0 (lanes 0-15)
    // and K=k0+2 (lanes 16-31); vgpr1 holds K=k0+1 / k0+3.
    v2f breg[32];
#pragma unroll
    for (int kk = 0; kk < 32; ++kk) {
        int k = (kk << 2) + (halfSel << 1);
        breg[kk].x = W[(size_t)k * D_DIM + col];
        breg[kk].y = W[(size_t)(k + 1) * D_DIM + col];
    }
    const float bn = bias ? bias[col] : 0.0f;

    const int nTiles = nRows >> 4;
    const int stride = gridDim.x;

#if USE_TDM
    const unsigned ldsBase = (unsigned)(uintptr_t)(&Atile[0][0][0]);
    const unsigned bufBytes = 16u * 132u * 4u;
    if (wave == 0 && blockIdx.x < nTiles)            // prefetch first tile -> buf 0
        tdm_load_tile(ldsBase, (unsigned long long)(uintptr_t)(A) +
                               (unsigned long long)blockIdx.x * 16ull * D_DIM * 4ull);
#endif

    int buf = 0;
    for (int tile = blockIdx.x; tile < nTiles; tile += stride) {
        const int m0 = tile << 4;
#if USE_TDM
        if (wave == 0) {
            int nxt = tile + stride;
            if (nxt < nTiles) {                      // prefetch next tile -> other buf
                tdm_load_tile(ldsBase + (unsigned)(buf ^ 1) * bufBytes,
                              (unsigned long long)(uintptr_t)(A) +
                              (unsigned long long)nxt * 16ull * D_DIM * 4ull);
                __builtin_amdgcn_s_wait_tensorcnt((short)1);  // current buf's TDM done
            } else {
                __builtin_amdgcn_s_wait_tensorcnt((short)0);
            }
        }
#else
        {   // cooperative staging fallback: 256 thr x 8 floats (pure copy)
            int r = threadIdx.x >> 4;
            int c = (threadIdx.x & 15) << 3;
            const float4* s4 = reinterpret_cast<const float4*>(A + (size_t)(m0 + r) * D_DIM + c);
            float4 v0 = s4[0], v1 = s4[1];
            float* dp = &Atile[buf][r][c];
            dp[0] = v0.x; dp[1] = v0.y; dp[2] = v0.z; dp[3] = v0.w;
            dp[4] = v1.x; dp[5] = v1.y; dp[6] = v1.z; dp[7] = v1.w;
        }
#endif
        __syncthreads();

        v8f acc = {0.f, 0.f, 0.f, 0.f, 0.f, 0.f, 0.f, 0.f};
#pragma unroll
        for (int kk = 0; kk < 32; ++kk) {
            int k = (kk << 2) + (halfSel << 1);
            v2f a;
            a.x = Atile[buf][lane16][k];
            a.y = Atile[buf][lane16][k + 1];
            acc = __builtin_amdgcn_wmma_f32_16x16x4_f32(
                false, a, false, breg[kk], (short)0, acc, false, false);
        }

        // C/D layout: VGPR r -> row m0 + r + 8*halfSel, col n0 + lane16
#pragma unroll
        for (int r = 0; r < 8; ++r) {
            int row = m0 + r + (halfSel << 3);
            float v = acc[r];
            if (rowscale) v *= rowscale[row];
            v += bn;
            if (postscale) v *= postscale[row];
            out[(size_t)row * D_DIM + col] = v;
        }
        __syncthreads();                             // all reads of `buf` done
        buf ^= 1;
    }
}

// ---------------------------------------------------------------------------
// BatchNorm column statistics (biased var), then fused BN+ReLU+residual
// ---------------------------------------------------------------------------
__global__ void col_stats(const float* __restrict__ t,
                          float* __restrict__ sums, float* __restrict__ sq, int nRows) {
    const int col = threadIdx.x;                      // blockDim = 128
    long long r0 = (long long)blockIdx.x * 256;
    long long r1 = r0 + 256; if (r1 > nRows) r1 = nRows;
    float s = 0.f, q = 0.f;
    for (long long r = r0; r < r1; ++r) {
        float v = t[r * D_DIM + col];
        s += v; q += v * v;
    }
    unsafeAtomicAdd(&sums[col], s);
    unsafeAtomicAdd(&sq[col], q);
}

__global__ void finalize_stats(const float* __restrict__ sums, const float* __restrict__ sq,
                               float* __restrict__ mu, float* __restrict__ istd, float invN) {
    int c = threadIdx.x;                              // 128 threads
    float m = sums[c] * invN;
    float v = sq[c] * invN - m * m;
    mu[c]   = m;
    istd[c] = rsqrtf(v + BN_EPS);
}

__global__ void bn_relu_residual(float* __restrict__ h, const float* __restrict__ t,
                                 const float* __restrict__ mu, const float* __restrict__ istd,
                                 const float* __restrict__ gamma, const float* __restrict__ beta,
                                 long long n) {
    long long i = (long long)blockIdx.x * blockDim.x + threadIdx.x;
    if (i >= n) return;
    int c = (int)(i & (D_DIM - 1));
    float v = (t[i] - mu[c]) * istd[c] * gamma[c] + beta[c];
    v = fmaxf(v, 0.0f);
    h[i] = h[i] + v;                                  // residual
}

// ---------------------------------------------------------------------------
// readout: segment mean over graphs, then hg @ W_out
// ---------------------------------------------------------------------------
__global__ void graph_count(const int* __restrict__ gid, float* __restrict__ cnt, int n) {
    int i = blockIdx.x * blockDim.x + threadIdx.x;
    if (i < n) unsafeAtomicAdd(&cnt[gid[i]], 1.0f);
}

__global__ void seg_accumulate(const float* __restrict__ h, const int* __restrict__ gid,
                               float* __restrict__ hg, int nNodes) {
    long long tid = (long long)blockIdx.x * blockDim.x + threadIdx.x;
    int node = (int)(tid >> 5);
    if (node >= nNodes) return;
    int lane = (int)(tid & 31);
    int g = gid[node];
    const float4 v = *reinterpret_cast<const float4*>(h + (size_t)node * D_DIM + lane * 4);
    float* o = hg + (size_t)g * D_DIM + lane * 4;
    unsafeAtomicAdd(o + 0, v.x);
    unsafeAtomicAdd(o + 1, v.y);
    unsafeAtomicAdd(o + 2, v.z);
    unsafeAtomicAdd(o + 3, v.w);
}

__global__ void readout_mlp(const float* __restrict__ hg, const float* __restrict__ cnt,
                            const float* __restrict__ Wout, float* __restrict__ out) {
    int g = blockIdx.x;
    int c = threadIdx.x;
    if (c >= N_CLS) return;
    float icnt = 1.0f / fmaxf(cnt[g], 1.0f);
    float s = 0.0f;
    for (int k = 0; k < D_DIM; ++k)
        s += hg[(size_t)g * D_DIM + k] * Wout[k * N_CLS + c];
    out[(size_t)g * N_CLS + c] = s * icnt;
}

// ---------------------------------------------------------------------------
// host orchestration
// ---------------------------------------------------------------------------
static inline int cdiv_i(long long a, int b) { return (int)((a + b - 1) / b); }

extern "C" void kernel_launch(void* const* d_in, const int* in_sizes, int n_in,
                              void* d_out, int out_size, void* d_ws, size_t ws_size,
                              hipStream_t stream) {
    const float* nodes_feat = (const float*)d_in[0];
    // d_in[1] = edges_feat (unused)
    const float* snorm  = (const float*)d_in[2];     // nodes_num_norm_sqrt [N]
    const int*   src    = (const int*)d_in[3];
    const int*   dst    = (const int*)d_in[4];
    const int*   gids   = (const int*)d_in[5];
    const float* W_emb  = (const float*)d_in[6];
    const float* b_emb  = (const float*)d_in[7];
    const float* Ws     = (const float*)d_in[8];
    const float* bs     = (const float*)d_in[9];
    const float* gammas = (const float*)d_in[10];
    const float* betas  = (const float*)d_in[11];
    const float* W_out  = (const float*)d_in[12];
    float* out = (float*)d_out;

    const int N = in_sizes[0] / D_DIM;               // 100000
    const int E = in_sizes[3];                       // 1600000
    const int G = out_size / N_CLS;                  // 512
    const int L = in_sizes[8] / (D_DIM * D_DIM);     // 4
    const long long ND = (long long)N * D_DIM;

    // workspace carve-out
    char* wp = (char*)d_ws;
    size_t off = 0;
    auto take = [&](size_t nfloats) { float* p = (float*)(wp + off); off += nfloats * sizeof(float); return p; };
    float* h       = take(ND);
    float* t       = take(ND);
    float* agg     = take(ND);
    float* inv_out = take(N);
    float* inv_in  = take(N);
    float* colsum  = take(D_DIM);
    float* colsq   = take(D_DIM);
    float* mu      = take(D_DIM);
    float* istd    = take(D_DIM);
    float* hg      = take((size_t)G * D_DIM);
    float* cnt     = take(G);
    (void)ws_size; (void)n_in;

    // 1) symmetric degree norms
    fill_zero_f32<<<cdiv_i(N, 256), 256, 0, stream>>>(inv_out, N);
    fill_zero_f32<<<cdiv_i(N, 256), 256, 0, stream>>>(inv_in, N);
    count_deg<<<cdiv_i(E, 256), 256, 0, stream>>>(src, dst, inv_out, inv_in, E);
    inv_sqrt_deg<<<cdiv_i(N, 256), 256, 0, stream>>>(inv_out, inv_in, N);

    // 2) embedding: h = X @ W_emb + b_emb
    gemm128_wmma<<<1024, 256, 0, stream>>>(nodes_feat, W_emb, b_emb,
                                           nullptr, nullptr, h, N);

    // 3) GCN layers
    for (int l = 0; l < L; ++l) {
        fill_zero_f32<<<cdiv_i(ND, 256), 256, 0, stream>>>(agg, ND);
        gather_scatter<<<cdiv_i((long long)E * 32, 256), 256, 0, stream>>>(
            h, inv_out, src, dst, agg, E);
        // t = ((agg @ Ws[l]) * inv_in + bs[l]) * snorm
        gemm128_wmma<<<1024, 256, 0, stream>>>(agg, Ws + (size_t)l * D_DIM * D_DIM,
                                               bs + (size_t)l * D_DIM,
                                               inv_in, snorm, t, N);
        fill_zero_f32<<<1, 256, 0, stream>>>(colsum, 2 * D_DIM);  // colsum+colsq contiguous
        col_stats<<<cdiv_i(N, 256), 128, 0, stream>>>(t, colsum, colsq, N);
        finalize_stats<<<1, 128, 0, stream>>>(colsum, colsq, mu, istd, 1.0f / (float)N);
        bn_relu_residual<<<cdiv_i(ND, 256), 256, 0, stream>>>(
            h, t, mu, istd, gammas + (size_t)l * D_DIM, betas + (size_t)l * D_DIM, ND);
    }

    // 4) readout: mean over graphs, then @ W_out
    fill_zero_f32<<<cdiv_i((long long)G * D_DIM, 256), 256, 0, stream>>>(hg, (long long)G * D_DIM);
    fill_zero_f32<<<1, 256, 0, stream>>>(cnt, G);
    graph_count<<<cdiv_i(N, 256), 256, 0, stream>>>(gids, cnt, N);
    seg_accumulate<<<cdiv_i((long long)N * 32, 256), 256, 0, stream>>>(h, gids, hg, N);
    readout_mlp<<<G, 32, 0, stream>>>(hg, cnt, W_out, out);
}